// GCN_21234318311432
// MI455X (gfx1250) — compile-verified
//
#include <hip/hip_runtime.h>

#define NNODES 50000
#define NEDGES 1600000
#define NGRAPHS 256
#define HID 128
#define OUTC 10

typedef float v2f __attribute__((ext_vector_type(2)));
typedef float v8f __attribute__((ext_vector_type(8)));

// ---------------------------------------------------------------------------
// Degree / normalization
// ---------------------------------------------------------------------------
__global__ void __launch_bounds__(256) k_init_deg(float* deg, int n) {
    int i = blockIdx.x * 256 + threadIdx.x;
    if (i < n) deg[i] = 1.0f;   // self-loop weight 1
}

__global__ void __launch_bounds__(256) k_edge_deg(const int* __restrict__ dst,
                                                  const float* __restrict__ ew,
                                                  float* deg, int ne) {
    int e = blockIdx.x * 256 + threadIdx.x;
    if (e < ne) unsafeAtomicAdd(&deg[dst[e]], ew[e]);
}

__global__ void __launch_bounds__(256) k_rsqrt(float* d, int n) {
    int i = blockIdx.x * 256 + threadIdx.x;
    if (i < n) {
        float v = d[i];
        d[i] = (v > 0.0f) ? rsqrtf(v) : 0.0f;
    }
}

// ---------------------------------------------------------------------------
// WMMA f32 GEMM: H[rows x 128] = X[rows x 128] @ W[128 x 128]
// One wave per 16x16 output tile, K walked in steps of 4 via
// V_WMMA_F32_16X16X4_F32 (full fp32 precision — matches reference).
// rows must be a multiple of 16 (50000 = 3125*16, 256 = 16*16).
// ---------------------------------------------------------------------------
__global__ void __launch_bounds__(256) k_gemm128_wmma(const float* __restrict__ X,
                                                      const float* __restrict__ W,
                                                      float* __restrict__ H,
                                                      int rows) {
    int wave  = (blockIdx.x * 256 + threadIdx.x) >> 5;
    int lane  = threadIdx.x & 31;
    int tileM = wave >> 3;        // 8 col-tiles (128/16)
    int tileN = wave & 7;
    if (tileM * 16 >= rows) return;
    int m0   = tileM * 16;
    int n0   = tileN * 16;
    int lmod = lane & 15;
    int lhi  = lane >> 4;         // 0: K,K+1   1: K+2,K+3

    v8f acc = {};
    const float* xrow = X + (size_t)(m0 + lmod) * HID;
    #pragma unroll 4
    for (int k = 0; k < HID; k += 4) {
        // A (16x4 f32): lanes 0-15 rows M, vgpr0=K, vgpr1=K+1; lanes 16-31: K+2/K+3
        v2f a;
        a.x = xrow[k + 2 * lhi + 0];
        a.y = xrow[k + 2 * lhi + 1];
        // B (4x16 f32): row-striped across lanes, mirrored K split
        v2f b;
        b.x = W[(size_t)(k + 2 * lhi + 0) * HID + n0 + lmod];
        b.y = W[(size_t)(k + 2 * lhi + 1) * HID + n0 + lmod];
        acc = __builtin_amdgcn_wmma_f32_16x16x4_f32(
                false, a, false, b, (short)0, acc, false, false);
    }
    // C/D layout: lane<16 -> N=lane, M=m0+vgpr; lane>=16 -> N=lane-16, M=m0+8+vgpr
    float* hbase = H + (size_t)(m0 + lhi * 8) * HID + n0 + lmod;
    #pragma unroll
    for (int r = 0; r < 8; ++r)
        hbase[(size_t)r * HID] = acc[r];
}

// ---------------------------------------------------------------------------
// out[i][:] = h[i][:] * dinv[i]^2   (self-loop term initializes accumulator)
// one wave per node, float4 per lane (32*16B = 128 floats)
// ---------------------------------------------------------------------------
__global__ void __launch_bounds__(256) k_selfloop_init(const float* __restrict__ h,
                                                       const float* __restrict__ dinv,
                                                       float* __restrict__ out,
                                                       int nnodes) {
    int gid  = blockIdx.x * 256 + threadIdx.x;
    int node = gid >> 5;
    int lane = gid & 31;
    if (node >= nnodes) return;
    float di = dinv[node];
    float w  = di * di;
    float4 v = ((const float4*)(h + (size_t)node * HID))[lane];
    float4 r = make_float4(v.x * w, v.y * w, v.z * w, v.w * w);
    ((float4*)(out + (size_t)node * HID))[lane] = r;
}

// ---------------------------------------------------------------------------
// Edge scatter: one wave per edge; lane gathers float4 of h[src],
// scales by norm = dinv[s]*dinv[d]*ew, atomically adds into out[dst].
// h and out are L2-resident (25.6 MB each << 192 MB L2).
// ---------------------------------------------------------------------------
__global__ void __launch_bounds__(256) k_edge_scatter(const int* __restrict__ src,
                                                      const int* __restrict__ dst,
                                                      const float* __restrict__ ew,
                                                      const float* __restrict__ dinv,
                                                      const float* __restrict__ h,
                                                      float* __restrict__ out,
                                                      int ne) {
    int gid  = blockIdx.x * 256 + threadIdx.x;
    int e    = gid >> 5;
    int lane = gid & 31;
    if (e >= ne) return;
    int s = src[e];
    int d = dst[e];
    float norm = dinv[s] * dinv[d] * ew[e];
    float4 v = ((const float4*)(h + (size_t)s * HID))[lane];
    float* op = out + (size_t)d * HID + lane * 4;
    unsafeAtomicAdd(op + 0, v.x * norm);
    unsafeAtomicAdd(op + 1, v.y * norm);
    unsafeAtomicAdd(op + 2, v.z * norm);
    unsafeAtomicAdd(op + 3, v.w * norm);
}

// ---------------------------------------------------------------------------
// out[i][c] += b[c]; optional ReLU
// ---------------------------------------------------------------------------
__global__ void __launch_bounds__(256) k_bias_act(float* __restrict__ out,
                                                  const float* __restrict__ b,
                                                  int n, int relu) {
    int i = blockIdx.x * 256 + threadIdx.x;
    if (i >= n) return;
    float v = out[i] + b[i & (HID - 1)];
    if (relu) v = fmaxf(v, 0.0f);
    out[i] = v;
}

// ---------------------------------------------------------------------------
// Mean-pool over graphs
// ---------------------------------------------------------------------------
__global__ void __launch_bounds__(256) k_pool_zero(float* emb, float* cnt) {
    int i = blockIdx.x * 256 + threadIdx.x;
    if (i < NGRAPHS * HID) emb[i] = 0.0f;
    if (i < NGRAPHS) cnt[i] = 0.0f;
}

__global__ void __launch_bounds__(256) k_pool_scatter(const float* __restrict__ h,
                                                      const int* __restrict__ batch,
                                                      float* emb, float* cnt,
                                                      int nnodes) {
    int gid  = blockIdx.x * 256 + threadIdx.x;
    int node = gid >> 5;
    int lane = gid & 31;
    if (node >= nnodes) return;
    int g = batch[node];
    float4 v = ((const float4*)(h + (size_t)node * HID))[lane];
    float* ep = emb + (size_t)g * HID + lane * 4;
    unsafeAtomicAdd(ep + 0, v.x);
    unsafeAtomicAdd(ep + 1, v.y);
    unsafeAtomicAdd(ep + 2, v.z);
    unsafeAtomicAdd(ep + 3, v.w);
    if (lane == 0) unsafeAtomicAdd(&cnt[g], 1.0f);
}

__global__ void __launch_bounds__(256) k_pool_div(float* emb, const float* cnt) {
    int i = blockIdx.x * 256 + threadIdx.x;
    if (i >= NGRAPHS * HID) return;
    emb[i] = emb[i] / fmaxf(cnt[i >> 7], 1.0f);
}

// ---------------------------------------------------------------------------
// Final classifier: out[g][o] = lb2[o] + sum_c (emb2[g][c] + lb1[c]) * lw2[c][o]
// (lb1 folded into the dot product; emb2 = emb @ lw1 via the WMMA GEMM)
// ---------------------------------------------------------------------------
__global__ void __launch_bounds__(256) k_final(const float* __restrict__ emb2,
                                               const float* __restrict__ lb1,
                                               const float* __restrict__ lw2,
                                               const float* __restrict__ lb2,
                                               float* __restrict__ out) {
    int i = blockIdx.x * 256 + threadIdx.x;
    if (i >= NGRAPHS * OUTC) return;
    int g = i / OUTC;
    int o = i % OUTC;
    float acc = lb2[o];
    const float* er = emb2 + (size_t)g * HID;
    #pragma unroll 8
    for (int c = 0; c < HID; ++c)
        acc = fmaf(er[c] + lb1[c], lw2[(size_t)c * OUTC + o], acc);
    out[i] = acc;
}

// ---------------------------------------------------------------------------
extern "C" void kernel_launch(void* const* d_in, const int* in_sizes, int n_in,
                              void* d_out, int out_size, void* d_ws, size_t ws_size,
                              hipStream_t stream) {
    const float* x   = (const float*)d_in[0];
    const int*   ei  = (const int*)d_in[1];          // [2, E] flat
    const float* ew  = (const float*)d_in[2];
    const int*   bat = (const int*)d_in[3];
    const float* W1  = (const float*)d_in[4];
    const float* b1  = (const float*)d_in[5];
    const float* W2  = (const float*)d_in[6];
    const float* b2  = (const float*)d_in[7];
    const float* W3  = (const float*)d_in[8];
    const float* b3  = (const float*)d_in[9];
    const float* lw1 = (const float*)d_in[10];
    const float* lb1 = (const float*)d_in[11];
    const float* lw2 = (const float*)d_in[12];
    const float* lb2 = (const float*)d_in[13];
    float* out = (float*)d_out;

    const int* src = ei;
    const int* dst = ei + NEDGES;

    // workspace layout (floats)
    float* ws   = (float*)d_ws;
    float* dinv = ws;                                   // NNODES (also deg)
    float* hbuf = ws + ((NNODES + 63) & ~63);           // NNODES*HID
    float* abuf = hbuf + (size_t)NNODES * HID;          // NNODES*HID
    float* emb  = abuf + (size_t)NNODES * HID;          // NGRAPHS*HID
    float* cnt  = emb + NGRAPHS * HID;                  // NGRAPHS
    float* emb2 = cnt + NGRAPHS;                        // NGRAPHS*HID

    const int nodeBlk  = (NNODES + 255) / 256;
    const int edgeBlk  = (NEDGES + 255) / 256;
    const int nodeWave = (NNODES * 32 + 255) / 256;     // wave-per-node grids
    const int edgeWave = (NEDGES * 32 + 255) / 256;     // wave-per-edge grids
    const int featBlk  = (NNODES * HID + 255) / 256;

    // --- degree normalization (shared across all three layers) ---
    k_init_deg<<<nodeBlk, 256, 0, stream>>>(dinv, NNODES);
    k_edge_deg<<<edgeBlk, 256, 0, stream>>>(dst, ew, dinv, NEDGES);
    k_rsqrt<<<nodeBlk, 256, 0, stream>>>(dinv, NNODES);

    // --- three GCN layers ---
    const float* layer_in[3] = { x, abuf, abuf };
    const float* layer_W[3]  = { W1, W2, W3 };
    const float* layer_b[3]  = { b1, b2, b3 };
    for (int l = 0; l < 3; ++l) {
        k_gemm128_wmma<<<NNODES / 16, 256, 0, stream>>>(layer_in[l], layer_W[l],
                                                        hbuf, NNODES);
        k_selfloop_init<<<nodeWave, 256, 0, stream>>>(hbuf, dinv, abuf, NNODES);
        k_edge_scatter<<<edgeWave, 256, 0, stream>>>(src, dst, ew, dinv,
                                                     hbuf, abuf, NEDGES);
        k_bias_act<<<featBlk, 256, 0, stream>>>(abuf, layer_b[l],
                                                NNODES * HID, l < 2 ? 1 : 0);
    }

    // --- global mean pool ---
    k_pool_zero<<<(NGRAPHS * HID + 255) / 256, 256, 0, stream>>>(emb, cnt);
    k_pool_scatter<<<nodeWave, 256, 0, stream>>>(abuf, bat, emb, cnt, NNODES);
    k_pool_div<<<(NGRAPHS * HID + 255) / 256, 256, 0, stream>>>(emb, cnt);

    // --- classifier: emb2 = emb @ lw1 (bias folded into final) ---
    k_gemm128_wmma<<<NGRAPHS / 16, 256, 0, stream>>>(emb, lw1, emb2, NGRAPHS);
    k_final<<<(NGRAPHS * OUTC + 255) / 256, 256, 0, stream>>>(emb2, lb1, lw2,
                                                              lb2, out);
}